// GCN_37726992728419
// MI455X (gfx1250) — compile-verified
//
#include <hip/hip_runtime.h>
#include <cstdint>
#include <cstddef>

typedef __attribute__((ext_vector_type(2))) float v2f;
typedef __attribute__((ext_vector_type(8))) float v8f;

#define F_IN   256
#define F_H    128
#define F_OUT  8
#define NGRAPH 64

// ---------------------------------------------------------------------------
// 1) Degree accumulation (float atomics; counts are small integers, exact)
// ---------------------------------------------------------------------------
__global__ void degree_kernel(const int* __restrict__ src, const int* __restrict__ dst,
                              float* __restrict__ deg_out, float* __restrict__ deg_in, int E) {
    int e = blockIdx.x * blockDim.x + threadIdx.x;
    if (e >= E) return;
    atomicAdd(&deg_out[src[e]], 1.0f);
    atomicAdd(&deg_in[dst[e]], 1.0f);
}

// deg -> 1/sqrt(max(deg,1)), in place
__global__ void inv_sqrt_kernel(float* __restrict__ deg_out, float* __restrict__ deg_in, int N) {
    int n = blockIdx.x * blockDim.x + threadIdx.x;
    if (n >= N) return;
    deg_out[n] = 1.0f / sqrtf(fmaxf(deg_out[n], 1.0f));
    deg_in[n]  = 1.0f / sqrtf(fmaxf(deg_in[n], 1.0f));
}

// ---------------------------------------------------------------------------
// 2) GEMM1: h[N,128] = (x * inv_out[:,None]) @ W1  via V_WMMA_F32_16X16X4_F32
//    Block = 256 threads = 8 waves. Block owns a 16-row strip; wave w owns
//    columns [16w,16w+16). A-strip (16x256, pre-scaled) staged in LDS once.
//    Uniform (scalar) bounds fast-path: only the final partial tile takes the
//    guarded slow path, so the hot loop + stores are branch-free.
// ---------------------------------------------------------------------------
__global__ void __launch_bounds__(256)
gemm1_wmma(const float* __restrict__ x, const float* __restrict__ W1,
           const float* __restrict__ inv_out, float* __restrict__ h, int N) {
    // stride 260: (r*260 + k) % 64 banks -> 4r+k, conflict-free for 16 rows
    __shared__ float As[16][260];
    const int  tid  = threadIdx.x;
    const int  base = blockIdx.x * 16;
    const bool full = (base + 16 <= N);   // uniform across the block

    {   // stage A = x[base : base+16, :] * inv_out, 16 threads per row
        int r   = tid >> 4;
        int c0  = (tid & 15) * 16;
        int row = base + r;
        if (full || row < N) {
            float s = inv_out[row];
            const float4* xr = (const float4*)(x + (size_t)row * F_IN);
            #pragma unroll
            for (int i = 0; i < 4; ++i) {
                float4 v = xr[(c0 >> 2) + i];
                As[r][c0 + i * 4 + 0] = v.x * s;
                As[r][c0 + i * 4 + 1] = v.y * s;
                As[r][c0 + i * 4 + 2] = v.z * s;
                As[r][c0 + i * 4 + 3] = v.w * s;
            }
        } else {
            #pragma unroll
            for (int i = 0; i < 16; ++i) As[r][c0 + i] = 0.0f;
        }
    }
    __syncthreads();

    const int wave = tid >> 5;
    const int lane = tid & 31;
    const int l    = lane & 15;
    const int koff = (lane >> 4) * 2;   // lanes 16-31 hold K+2,K+3 (ISA 7.12.2)
    const int col0 = wave * 16;

    v8f acc = {};
    #pragma unroll 4
    for (int k0 = 0; k0 < F_IN; k0 += 4) {
        v2f a, b;
        a[0] = As[l][k0 + koff];
        a[1] = As[l][k0 + koff + 1];
        b[0] = W1[(size_t)(k0 + koff)     * F_H + col0 + l];
        b[1] = W1[(size_t)(k0 + koff + 1) * F_H + col0 + l];
        acc = __builtin_amdgcn_wmma_f32_16x16x4_f32(false, a, false, b,
                                                    (short)0, acc, false, false);
    }

    // D layout: VGPR j -> row M=j (lanes 0-15) / M=j+8 (lanes 16-31)
    const int mbase = (lane >> 4) * 8;
    float* hp = h + (size_t)(base + mbase) * F_H + col0 + l;
    if (full) {
        #pragma unroll
        for (int j = 0; j < 8; ++j) hp[(size_t)j * F_H] = acc[j];
    } else {
        #pragma unroll
        for (int j = 0; j < 8; ++j)
            if (base + mbase + j < N) hp[(size_t)j * F_H] = acc[j];
    }
}

// ---------------------------------------------------------------------------
// 3) Edge scatter, layer 1: agg1[dst] += h[src]   (one wave32 per edge,
//    4 features per lane). h and agg1 are L2-resident (51.2 MB each,
//    192 MB L2) so gathers + atomics stay off HBM.
// ---------------------------------------------------------------------------
__global__ void __launch_bounds__(256)
scatter1_kernel(const float* __restrict__ h, const int* __restrict__ src,
                const int* __restrict__ dst, float* __restrict__ agg1, int E) {
    int wid  = (blockIdx.x * blockDim.x + threadIdx.x) >> 5;
    int lane = threadIdx.x & 31;
    if (wid >= E) return;
    int s = src[wid], d = dst[wid];
    float4 v = *(const float4*)(h + (size_t)s * F_H + lane * 4);
    float* p = agg1 + (size_t)d * F_H + lane * 4;
    atomicAdd(p + 0, v.x);
    atomicAdd(p + 1, v.y);
    atomicAdd(p + 2, v.z);
    atomicAdd(p + 3, v.w);
}

// ---------------------------------------------------------------------------
// 4) h1s = relu(agg1 * inv_in + b1) * inv_out   (pre-scaled for layer 2)
//    One thread per float4 (N*32 threads).
// ---------------------------------------------------------------------------
__global__ void __launch_bounds__(256)
post1_kernel(const float* __restrict__ agg1, const float* __restrict__ inv_in,
             const float* __restrict__ inv_out, const float* __restrict__ b1,
             float* __restrict__ h1s, int N) {
    int t = blockIdx.x * blockDim.x + threadIdx.x;
    if (t >= N * (F_H / 4)) return;
    int n  = t >> 5;            // F_H/4 == 32
    int f4 = t & 31;
    float wi = inv_in[n];
    float wo = inv_out[n];
    float4 v  = ((const float4*)agg1)[(size_t)n * 32 + f4];
    float4 bb = ((const float4*)b1)[f4];
    v.x = fmaxf(v.x * wi + bb.x, 0.0f) * wo;
    v.y = fmaxf(v.y * wi + bb.y, 0.0f) * wo;
    v.z = fmaxf(v.z * wi + bb.z, 0.0f) * wo;
    v.w = fmaxf(v.w * wi + bb.w, 0.0f) * wo;
    ((float4*)h1s)[(size_t)n * 32 + f4] = v;
}

// ---------------------------------------------------------------------------
// 5) GEMM2: h2[N,8] = h1s @ W2. Tiny K/N -> VALU, W2 staged in LDS (broadcast
//    reads, conflict-free). One thread per node reads its row once.
// ---------------------------------------------------------------------------
__global__ void __launch_bounds__(256)
gemm2_kernel(const float* __restrict__ h1s, const float* __restrict__ W2,
             float* __restrict__ h2, int N) {
    __shared__ float w2s[F_H * F_OUT];   // 4 KB
    int tid = threadIdx.x;
    for (int i = tid; i < F_H * F_OUT; i += 256) w2s[i] = W2[i];
    __syncthreads();
    int n = blockIdx.x * blockDim.x + tid;
    if (n >= N) return;
    float acc[F_OUT] = {};
    const float4* row = (const float4*)(h1s + (size_t)n * F_H);
    #pragma unroll 4
    for (int k4 = 0; k4 < F_H / 4; ++k4) {
        float4 v = row[k4];
        int k = k4 * 4;
        #pragma unroll
        for (int o = 0; o < F_OUT; ++o) {
            acc[o] += v.x * w2s[(k + 0) * F_OUT + o]
                    + v.y * w2s[(k + 1) * F_OUT + o]
                    + v.z * w2s[(k + 2) * F_OUT + o]
                    + v.w * w2s[(k + 3) * F_OUT + o];
        }
    }
    #pragma unroll
    for (int o = 0; o < F_OUT; ++o) h2[(size_t)n * F_OUT + o] = acc[o];
}

// ---------------------------------------------------------------------------
// 6) Edge scatter, layer 2: agg2[dst] += h2[src]  (one thread per edge, 8 f32)
// ---------------------------------------------------------------------------
__global__ void __launch_bounds__(256)
scatter2_kernel(const float* __restrict__ h2, const int* __restrict__ src,
                const int* __restrict__ dst, float* __restrict__ agg2, int E) {
    int e = blockIdx.x * blockDim.x + threadIdx.x;
    if (e >= E) return;
    int s = src[e], d = dst[e];
    float4 a = *(const float4*)(h2 + (size_t)s * F_OUT);
    float4 b = *(const float4*)(h2 + (size_t)s * F_OUT + 4);
    float* p = agg2 + (size_t)d * F_OUT;
    atomicAdd(p + 0, a.x); atomicAdd(p + 1, a.y);
    atomicAdd(p + 2, a.z); atomicAdd(p + 3, a.w);
    atomicAdd(p + 4, b.x); atomicAdd(p + 5, b.y);
    atomicAdd(p + 6, b.z); atomicAdd(p + 7, b.w);
}

// ---------------------------------------------------------------------------
// 7) Per-graph sums via block-local LDS bins, then few global atomics
// ---------------------------------------------------------------------------
__global__ void __launch_bounds__(256)
graph_reduce_kernel(const float* __restrict__ agg2, const float* __restrict__ inv_in,
                    const float* __restrict__ b2, const int* __restrict__ gid,
                    float* __restrict__ gsums, float* __restrict__ gcnt, int N) {
    __shared__ float s[NGRAPH * F_OUT];
    __shared__ float c[NGRAPH];
    int tid = threadIdx.x;
    for (int i = tid; i < NGRAPH * F_OUT; i += 256) s[i] = 0.0f;
    if (tid < NGRAPH) c[tid] = 0.0f;
    __syncthreads();
    int n = blockIdx.x * blockDim.x + tid;
    if (n < N) {
        int g = gid[n];
        float w = inv_in[n];
        atomicAdd(&c[g], 1.0f);
        #pragma unroll
        for (int o = 0; o < F_OUT; ++o) {
            float v = agg2[(size_t)n * F_OUT + o] * w + b2[o];
            atomicAdd(&s[g * F_OUT + o], v);
        }
    }
    __syncthreads();
    for (int i = tid; i < NGRAPH * F_OUT; i += 256) atomicAdd(&gsums[i], s[i]);
    if (tid < NGRAPH) atomicAdd(&gcnt[tid], c[tid]);
}

__global__ void finalize_kernel(const float* __restrict__ gsums,
                                const float* __restrict__ gcnt,
                                float* __restrict__ out) {
    int i = threadIdx.x;                       // 512 threads
    if (i < NGRAPH * F_OUT)
        out[i] = gsums[i] / fmaxf(gcnt[i >> 3], 1.0f);
}

// ---------------------------------------------------------------------------
// Launcher
// ---------------------------------------------------------------------------
extern "C" void kernel_launch(void* const* d_in, const int* in_sizes, int n_in,
                              void* d_out, int out_size, void* d_ws, size_t ws_size,
                              hipStream_t stream) {
    const float* x   = (const float*)d_in[0];
    const float* W1  = (const float*)d_in[1];
    const float* b1  = (const float*)d_in[2];
    const float* W2  = (const float*)d_in[3];
    const float* b2  = (const float*)d_in[4];
    const int*   src = (const int*)d_in[5];
    const int*   dst = (const int*)d_in[6];
    const int*   gid = (const int*)d_in[7];
    float* out = (float*)d_out;

    const int E = in_sizes[5];
    const int N = in_sizes[7];

    // Workspace layout (floats)
    float* ws      = (float*)d_ws;
    float* inv_out = ws;                               // N   (deg_out -> rsqrt)
    float* inv_in  = inv_out + (size_t)N;              // N
    float* h       = inv_in + (size_t)N;               // N*128 (reused as h1s)
    float* agg1    = h + (size_t)N * F_H;              // N*128
    float* h2      = agg1 + (size_t)N * F_H;           // N*8
    float* agg2    = h2 + (size_t)N * F_OUT;           // N*8
    float* gsums   = agg2 + (size_t)N * F_OUT;         // 512
    float* gcnt    = gsums + NGRAPH * F_OUT;           // 64

    // Zero accumulators (memset nodes are graph-capturable)
    hipMemsetAsync(inv_out, 0, (size_t)2 * N * sizeof(float), stream);
    hipMemsetAsync(agg1,    0, (size_t)N * F_H * sizeof(float), stream);
    hipMemsetAsync(agg2,    0, (size_t)N * F_OUT * sizeof(float), stream);
    hipMemsetAsync(gsums,   0, (NGRAPH * F_OUT + NGRAPH) * sizeof(float), stream);

    const int B = 256;
    degree_kernel<<<(E + B - 1) / B, B, 0, stream>>>(src, dst, inv_out, inv_in, E);
    inv_sqrt_kernel<<<(N + B - 1) / B, B, 0, stream>>>(inv_out, inv_in, N);

    gemm1_wmma<<<(N + 15) / 16, B, 0, stream>>>(x, W1, inv_out, h, N);

    {   // one wave per edge
        long long thr = (long long)E * 32;
        scatter1_kernel<<<(unsigned)((thr + B - 1) / B), B, 0, stream>>>(h, src, dst, agg1, E);
    }

    {   // fused bias/relu/scale, float4 granularity; overwrites h with h1s
        long long thr = (long long)N * (F_H / 4);
        post1_kernel<<<(unsigned)((thr + B - 1) / B), B, 0, stream>>>(agg1, inv_in, inv_out, b1, h, N);
    }

    gemm2_kernel<<<(N + B - 1) / B, B, 0, stream>>>(h, W2, h2, N);

    scatter2_kernel<<<(E + B - 1) / B, B, 0, stream>>>(h2, src, dst, agg2, E);

    graph_reduce_kernel<<<(N + B - 1) / B, B, 0, stream>>>(agg2, inv_in, b2, gid, gsums, gcnt, N);

    finalize_kernel<<<1, 512, 0, stream>>>(gsums, gcnt, out);
}